// GIN_16518444220924
// MI455X (gfx1250) — compile-verified
//
#include <hip/hip_runtime.h>
#include <hip/hip_bf16.h>

typedef __attribute__((ext_vector_type(2))) float v2f;
typedef __attribute__((ext_vector_type(8))) float v8f;

#define D 128
#define TILE_M 16
#define LDS_STRIDE 132   // 128 + 4 pad: spreads WMMA-layout column reads across banks

// ---------------------------------------------------------------------------
// Native fp32 global atomic add (global_atomic_add_f32 on gfx1250)
// ---------------------------------------------------------------------------
__device__ __forceinline__ void atomic_add_f32(float* p, float v) {
    unsafeAtomicAdd(p, v);
}

// ---------------------------------------------------------------------------
// CDNA5 async global->LDS copy (16B per lane, tracked with ASYNCcnt).
// GVS mode: 64-bit SGPR base + per-lane 32-bit byte offset; LDS byte address
// in a VGPR.  Bypasses VGPRs entirely.
// ---------------------------------------------------------------------------
__device__ __forceinline__ void async_load_b128(unsigned lds_byte_addr,
                                                unsigned gmem_byte_off,
                                                const void* base) {
    asm volatile("global_load_async_to_lds_b128 %0, %1, %2"
                 :
                 : "v"(lds_byte_addr), "v"(gmem_byte_off), "s"(base)
                 : "memory");
}

__device__ __forceinline__ void wait_asynccnt0() {
    asm volatile("s_wait_asynccnt 0x0" ::: "memory");
}

// ---------------------------------------------------------------------------
// Pack a 128x128 row-major weight W (used as x @ W^T) into the WMMA B-operand
// lane layout for V_WMMA_F32_16X16X4_F32.
// B[k][n] = W[n][k].  Layout per (col-tile w, k-step s):
//   lane L (L<16):  {B[4s+0][w*16+L], B[4s+1][w*16+L]}
//   lane L+16:      {B[4s+2][w*16+L], B[4s+3][w*16+L]}
// Flat: P[((w*32 + s)*32 + lane)] as float2.  8 tiles * 32 steps * 32 lanes.
// ---------------------------------------------------------------------------
__global__ __launch_bounds__(256) void pack_w_kernel(const float* __restrict__ W,
                                                     float* __restrict__ P) {
    int tid = blockIdx.x * 256 + threadIdx.x;   // 0 .. 8191
    if (tid >= 8 * 32 * 32) return;
    int w    = tid >> 10;
    int rem  = tid & 1023;
    int s    = rem >> 5;
    int lane = rem & 31;
    int half = lane >> 4;
    int m    = lane & 15;
    int n    = w * 16 + m;
    int k    = 4 * s + 2 * half;
    float2 v;
    v.x = W[n * D + k];
    v.y = W[n * D + k + 1];
    ((float2*)P)[tid] = v;
}

// ---------------------------------------------------------------------------
// Edge scatter: one wave32 per edge.  Lane l moves float4 at feature offset
// 4*l (whole 512B row coalesced), atomically accumulates into acc[dst].
// Feature table (51.2 MB) is resident in the 192 MB L2, so both the random
// gather and the scatter atomics run at L2 throughput.
// ---------------------------------------------------------------------------
__global__ __launch_bounds__(256) void scatter_kernel(const float* __restrict__ X,
                                                      const int* __restrict__ src,
                                                      const int* __restrict__ dst,
                                                      float* __restrict__ acc,
                                                      float* __restrict__ deg,
                                                      int n_edges) {
    int e    = blockIdx.x * 8 + (threadIdx.x >> 5);
    int lane = threadIdx.x & 31;
    if (e >= n_edges) return;
    int s = src[e];
    int d = dst[e];
    float4 v = ((const float4*)(X + (size_t)s * D))[lane];
    float* out = acc + (size_t)d * D + lane * 4;
    atomic_add_f32(out + 0, v.x);
    atomic_add_f32(out + 1, v.y);
    atomic_add_f32(out + 2, v.z);
    atomic_add_f32(out + 3, v.w);
    if (lane == 0) atomic_add_f32(deg + d, 1.0f);
}

// ---------------------------------------------------------------------------
// h0 = x + acc / max(deg, 1)   (in place into acc)
// ---------------------------------------------------------------------------
__global__ __launch_bounds__(256) void finish_mean_kernel(const float* __restrict__ X,
                                                          float* __restrict__ acc,
                                                          const float* __restrict__ deg,
                                                          int n) {
    int tid = blockIdx.x * 256 + threadIdx.x;
    if (tid >= n * D) return;
    int v = tid >> 7;
    float inv = 1.0f / fmaxf(deg[v], 1.0f);
    acc[tid] = X[tid] + acc[tid] * inv;
}

// ---------------------------------------------------------------------------
// Fused 2-linear MLP with WMMA fp32:
//   Y = relu( (X @ Wa^T + ba) @ Wb^T + bb )
// Block = 256 threads = 8 waves.  Each block: 16 rows x 128 cols.
// Wave w owns columns [16w, 16w+16).  K=128 -> 32 x V_WMMA_F32_16X16X4_F32.
// A tile staged into LDS via global_load_async_to_lds_b128 (no VGPR transit);
// the 16x128 intermediate tile stays in LDS (no global round-trip).
// ---------------------------------------------------------------------------
__global__ __launch_bounds__(256) void mlp2_wmma_kernel(const float* __restrict__ X,
                                                        const float* __restrict__ Pa,
                                                        const float* __restrict__ ba,
                                                        const float* __restrict__ Pb,
                                                        const float* __restrict__ bb,
                                                        float* __restrict__ Y,
                                                        int n) {
    __shared__ float a_lds[TILE_M * LDS_STRIDE];
    __shared__ float t_lds[TILE_M * LDS_STRIDE];

    const int row0 = blockIdx.x * TILE_M;
    const int tid  = threadIdx.x;
    const int wave = tid >> 5;     // 0..7 : column tile
    const int lane = tid & 31;
    const int half = lane >> 4;    // 0 or 1
    const int m    = lane & 15;

    // Async-stage the 16x128 A tile directly into LDS: 512 x b128, 2 per
    // thread.  Rows past n are clamped to n-1 (per-row outputs are
    // independent and out-of-range rows are never stored), keeping EXEC
    // uniform for the async ops and the WMMAs.
    #pragma unroll
    for (int i = 0; i < 2; ++i) {
        int idx4 = tid + i * 256;      // float4 index within tile
        int r    = idx4 >> 5;          // 32 float4 per row
        int c4   = idx4 & 31;
        int row  = row0 + r;
        if (row > n - 1) row = n - 1;
        unsigned goff = ((unsigned)row * D + (unsigned)c4 * 4) * sizeof(float);
        unsigned laddr = (unsigned)(size_t)(&a_lds[r * LDS_STRIDE + c4 * 4]);
        async_load_b128(laddr, goff, X);
    }
    wait_asynccnt0();
    __syncthreads();

    // ---- Stage 1: T = X @ Wa^T + ba ----
    v8f acc = {0.f, 0.f, 0.f, 0.f, 0.f, 0.f, 0.f, 0.f};
    {
        const float* bpack = Pa + (size_t)wave * 2048;   // 32 steps * 64 floats
        #pragma unroll
        for (int s = 0; s < 32; ++s) {
            int k0 = 4 * s;
            v2f a = *(const v2f*)(&a_lds[m * LDS_STRIDE + k0 + 2 * half]);
            v2f b = ((const v2f*)(bpack + s * 64))[lane];
            acc = __builtin_amdgcn_wmma_f32_16x16x4_f32(
                false, a, false, b, (short)0, acc, false, false);
        }
        float bias = ba[wave * 16 + m];
        #pragma unroll
        for (int j = 0; j < 8; ++j) {
            t_lds[(j + 8 * half) * LDS_STRIDE + wave * 16 + m] = acc[j] + bias;
        }
    }
    __syncthreads();

    // ---- Stage 2: Y = relu(T @ Wb^T + bb) ----
    acc = (v8f){0.f, 0.f, 0.f, 0.f, 0.f, 0.f, 0.f, 0.f};
    {
        const float* bpack = Pb + (size_t)wave * 2048;
        #pragma unroll
        for (int s = 0; s < 32; ++s) {
            int k0 = 4 * s;
            v2f a = *(const v2f*)(&t_lds[m * LDS_STRIDE + k0 + 2 * half]);
            v2f b = ((const v2f*)(bpack + s * 64))[lane];
            acc = __builtin_amdgcn_wmma_f32_16x16x4_f32(
                false, a, false, b, (short)0, acc, false, false);
        }
        float bias = bb[wave * 16 + m];
        #pragma unroll
        for (int j = 0; j < 8; ++j) {
            int row = row0 + j + 8 * half;
            if (row < n) {
                Y[(size_t)row * D + wave * 16 + m] = fmaxf(acc[j] + bias, 0.f);
            }
        }
    }
}

// ---------------------------------------------------------------------------
// Host-side orchestration
// ---------------------------------------------------------------------------
static inline size_t align_up(size_t x, size_t a) { return (x + a - 1) & ~(a - 1); }

extern "C" void kernel_launch(void* const* d_in, const int* in_sizes, int n_in,
                              void* d_out, int out_size, void* d_ws, size_t ws_size,
                              hipStream_t stream) {
    const float* feat = (const float*)d_in[0];
    const int*   src  = (const int*)d_in[1];
    const int*   dst  = (const int*)d_in[2];
    const float* W1a  = (const float*)d_in[3];
    const float* b1a  = (const float*)d_in[4];
    const float* W1b  = (const float*)d_in[5];
    const float* b1b  = (const float*)d_in[6];
    const float* W2a  = (const float*)d_in[7];
    const float* b2a  = (const float*)d_in[8];
    const float* W2b  = (const float*)d_in[9];
    const float* b2b  = (const float*)d_in[10];
    float*       out  = (float*)d_out;

    const int n       = in_sizes[0] / D;
    const int n_edges = in_sizes[1];

    const size_t featBytes = (size_t)n * D * sizeof(float);
    const size_t packBytes = (size_t)8 * 32 * 32 * 2 * sizeof(float);   // 64 KB

    char* ws = (char*)d_ws;
    float* bufA = (float*)ws;  ws += align_up(featBytes, 256);
    float* bufB = (float*)ws;  ws += align_up(featBytes, 256);
    float* deg  = (float*)ws;  ws += align_up((size_t)n * sizeof(float), 256);
    float* p1a  = (float*)ws;  ws += packBytes;
    float* p1b  = (float*)ws;  ws += packBytes;
    float* p2a  = (float*)ws;  ws += packBytes;
    float* p2b  = (float*)ws;  ws += packBytes;

    // Pack the 4 weight matrices into WMMA B-operand layout (once per launch,
    // deterministic, 64 KB each).
    pack_w_kernel<<<32, 256, 0, stream>>>(W1a, p1a);
    pack_w_kernel<<<32, 256, 0, stream>>>(W1b, p1b);
    pack_w_kernel<<<32, 256, 0, stream>>>(W2a, p2a);
    pack_w_kernel<<<32, 256, 0, stream>>>(W2b, p2b);

    const int scatterBlocks = (n_edges + 7) / 8;
    const int finishBlocks  = (n * D + 255) / 256;
    const int mlpBlocks     = (n + TILE_M - 1) / TILE_M;

    // ---- Layer 1 ----
    hipMemsetAsync(bufA, 0, featBytes, stream);
    hipMemsetAsync(deg, 0, (size_t)n * sizeof(float), stream);
    scatter_kernel<<<scatterBlocks, 256, 0, stream>>>(feat, src, dst, bufA, deg, n_edges);
    finish_mean_kernel<<<finishBlocks, 256, 0, stream>>>(feat, bufA, deg, n);   // h0 in bufA
    mlp2_wmma_kernel<<<mlpBlocks, 256, 0, stream>>>(bufA, p1a, b1a, p1b, b1b, bufB, n); // h1 in bufB

    // ---- Layer 2 ----
    hipMemsetAsync(bufA, 0, featBytes, stream);
    hipMemsetAsync(deg, 0, (size_t)n * sizeof(float), stream);
    scatter_kernel<<<scatterBlocks, 256, 0, stream>>>(bufB, src, dst, bufA, deg, n_edges);
    finish_mean_kernel<<<finishBlocks, 256, 0, stream>>>(bufB, bufA, deg, n);   // h2 in bufA
    mlp2_wmma_kernel<<<mlpBlocks, 256, 0, stream>>>(bufA, p2a, b2a, p2b, b2b, out, n);
}